// MLC_10660108828924
// MI455X (gfx1250) — compile-verified
//
#include <hip/hip_runtime.h>

// Problem constants (match reference)
#define BROWS   16384
#define DIN     2048
#define CLASSES 156
#define CTILES  10      // 160 padded classes = 10 tiles of 16
#define DSEM    512
#define TOPK    10

typedef float v2f __attribute__((ext_vector_type(2)));
typedef float v4f __attribute__((ext_vector_type(4)));
typedef float v8f __attribute__((ext_vector_type(8)));

#define NEG_HUGE (-3.402823466e38f)

// ---------------------------------------------------------------------------
// Kernel 1: fused fp32-WMMA GEMM + bias + softmax + top-K (indices to ws)
// One wave = 16 rows x 160 (padded) classes, fully register-resident.
// A is streamed with non-temporal loads (read-once, 134 MB) so W (reused by
// all 1024 waves) and E (reused by kernel 2) stay resident in the 192 MB L2.
// ---------------------------------------------------------------------------
__global__ __launch_bounds__(256)
void MLC_gemm_softmax_topk(const float* __restrict__ A,
                           const float* __restrict__ W,
                           const float* __restrict__ bias,
                           float* __restrict__ tags,      // [B, CLASSES]
                           int*   __restrict__ topk_idx)  // [B, TOPK]
{
    const int lane = threadIdx.x & 31;
    const int wave = threadIdx.x >> 5;
    const int half = lane >> 4;      // 0: lanes 0-15, 1: lanes 16-31
    const int ln   = lane & 15;
    const int row_base = (blockIdx.x * 8 + wave) * 16;

    v8f acc[CTILES];
#pragma unroll
    for (int t = 0; t < CTILES; ++t) acc[t] = (v8f)0.0f;

    // A fragment pointer: lane ln supplies row (row_base+ln);
    // half 0 -> K = k+0,k+1 ; half 1 -> K = k+2,k+3  (ISA 16x4 f32 A layout)
    const float* aRow = A + (size_t)(row_base + ln) * DIN + 2 * half;

    // Per-tile W pointers (column c = t*16 + ln), clamped for padded classes;
    // padded columns are overwritten with -inf before softmax.
    const float* wBase[CTILES];
#pragma unroll
    for (int t = 0; t < CTILES; ++t) {
        int c  = t * 16 + ln;
        int cc = (c < CLASSES) ? c : (CLASSES - 1);
        wBase[t] = W + (size_t)cc * DIN + 2 * half;
    }

    for (int k = 0; k < DIN; k += 4) {
        // A: read-once stream -> non-temporal (TH=NT), don't pollute L2
        v2f afrag = __builtin_nontemporal_load((const v2f*)(aRow + k));
#pragma unroll
        for (int t = 0; t < CTILES; ++t) {
            // W: hot in cache (every wave reads all of W) -> default RT policy
            v2f bfrag = *(const v2f*)(wBase[t] + k);
            acc[t] = __builtin_amdgcn_wmma_f32_16x16x4_f32(
                false, afrag, false, bfrag, (short)0, acc[t], false, false);
        }
    }

    // Bias + mask padded classes. acc[t][j] holds row (8*half + j), col t*16+ln.
    float bcol[CTILES];
    bool  valid[CTILES];
#pragma unroll
    for (int t = 0; t < CTILES; ++t) {
        int c    = t * 16 + ln;
        valid[t] = (c < CLASSES);
        bcol[t]  = valid[t] ? bias[c] : 0.0f;
    }
#pragma unroll
    for (int t = 0; t < CTILES; ++t) {
#pragma unroll
        for (int j = 0; j < 8; ++j)
            acc[t][j] = valid[t] ? (acc[t][j] + bcol[t]) : NEG_HUGE;
    }

    // Per-row softmax + top-K. Row (8*half + j) lives across the 16 lanes of
    // this half-wave; xor masks 1,2,4,8 stay inside the half in wave32.
#pragma unroll
    for (int j = 0; j < 8; ++j) {
        const int row = row_base + 8 * half + j;

        // row max
        float rmax = NEG_HUGE;
#pragma unroll
        for (int t = 0; t < CTILES; ++t) rmax = fmaxf(rmax, acc[t][j]);
#pragma unroll
        for (int m = 1; m <= 8; m <<= 1)
            rmax = fmaxf(rmax, __shfl_xor(rmax, m, 32));

        // exp in place (padded cols: exp(-inf) = 0) + row sum
        float rsum = 0.0f;
#pragma unroll
        for (int t = 0; t < CTILES; ++t) {
            float e = __expf(acc[t][j] - rmax);
            acc[t][j] = e;
            rsum += e;
        }
#pragma unroll
        for (int m = 1; m <= 8; m <<= 1)
            rsum += __shfl_xor(rsum, m, 32);
        const float inv = 1.0f / rsum;

        // store tags (softmax), write-once stream -> non-temporal stores;
        // keep unscaled e in acc for top-K (same ordering)
#pragma unroll
        for (int t = 0; t < CTILES; ++t) {
            int c = t * 16 + ln;
            if (c < CLASSES)
                __builtin_nontemporal_store(acc[t][j] * inv,
                                            tags + (size_t)row * CLASSES + c);
        }

        // iterative top-K with argmax reductions
#pragma unroll
        for (int kk = 0; kk < TOPK; ++kk) {
            float bv = NEG_HUGE;
            int   bt = 0;
#pragma unroll
            for (int t = 0; t < CTILES; ++t)
                if (acc[t][j] > bv) { bv = acc[t][j]; bt = t; }
            int bc = bt * 16 + ln;
#pragma unroll
            for (int m = 1; m <= 8; m <<= 1) {
                float ov = __shfl_xor(bv, m, 32);
                int   oc = __shfl_xor(bc, m, 32);
                if (ov > bv || (ov == bv && oc < bc)) { bv = ov; bc = oc; }
            }
            if (ln == 0)
                topk_idx[(size_t)row * TOPK + kk] = bc;
            // knock out the winner
            const int wt = bc >> 4;
            const int wn = bc & 15;
#pragma unroll
            for (int t = 0; t < CTILES; ++t)
                if (t == wt && ln == wn) acc[t][j] = -1.0f;
        }
    }
}

// ---------------------------------------------------------------------------
// Kernel 2: semantic-feature gather, one wave per (b,k) row, float4 streams.
// E (320 KB) stays L2-resident with default policy; the 335 MB output stream
// uses non-temporal stores (never re-read) so it doesn't rinse L2.
// ---------------------------------------------------------------------------
__global__ __launch_bounds__(256)
void MLC_gather_sem(const float* __restrict__ E,
                    const int*   __restrict__ topk_idx,
                    float*       __restrict__ sem)   // [B*TOPK, DSEM]
{
    const int wave_g = blockIdx.x * (blockDim.x >> 5) + (threadIdx.x >> 5);
    const int lane   = threadIdx.x & 31;
    if (wave_g >= BROWS * TOPK) return;

    const int cls = topk_idx[wave_g];
    const v4f* src = (const v4f*)(E + (size_t)cls * DSEM);
    v4f*       dst = (v4f*)(sem + (size_t)wave_g * DSEM);
#pragma unroll
    for (int i = 0; i < 4; ++i) {
        v4f v = src[lane + 32 * i];                       // L2-hot read
        __builtin_nontemporal_store(v, dst + lane + 32 * i); // NT write stream
    }
}

// ---------------------------------------------------------------------------
extern "C" void kernel_launch(void* const* d_in, const int* in_sizes, int n_in,
                              void* d_out, int out_size, void* d_ws, size_t ws_size,
                              hipStream_t stream)
{
    (void)in_sizes; (void)n_in; (void)out_size; (void)ws_size;

    const float* A    = (const float*)d_in[0];  // [B, DIN]
    const float* W    = (const float*)d_in[1];  // [CLASSES, DIN]
    const float* bias = (const float*)d_in[2];  // [CLASSES]
    const float* E    = (const float*)d_in[3];  // [CLASSES, DSEM]

    float* tags = (float*)d_out;                           // [B, CLASSES]
    float* sem  = (float*)d_out + (size_t)BROWS * CLASSES; // [B, TOPK, DSEM]
    int*   idx  = (int*)d_ws;                              // [B, TOPK]

    // Kernel 1: 1024 waves (B/16), 8 waves per block
    MLC_gemm_softmax_topk<<<BROWS / (16 * 8), 256, 0, stream>>>(A, W, bias, tags, idx);

    // Kernel 2: one wave per (b,k) row
    const int rows   = BROWS * TOPK;           // 163840
    const int blocks = rows / 8;               // 8 waves per block
    MLC_gather_sem<<<blocks, 256, 0, stream>>>(E, idx, sem);
}